// KNN_10548439679761
// MI455X (gfx1250) — compile-verified
//
#include <hip/hip_runtime.h>
#include <stdint.h>
#include <stddef.h>

// Problem constants (fixed by the reference)
#define N_NODES 8192
#define KNB     64
#define KM1     63
#define FDIM    128
#define ROWLEN  257                      // 2*F + 1
#define NK      (N_NODES * KNB)          // 524288 events
#define SAMPLES_ELEMS 134742016ULL       // NK * ROWLEN

#if defined(__has_builtin)
#  if __has_builtin(__builtin_amdgcn_global_load_async_to_lds_b128)
#    define HAVE_ASYNC_LDS 1
#  endif
#  if __has_builtin(__builtin_amdgcn_s_wait_asynccnt)
#    define HAVE_WAIT_ASYNC 1
#  endif
#endif

#ifdef HAVE_ASYNC_LDS
typedef int v4i_t __attribute__((__vector_size__(4 * sizeof(int))));
typedef __attribute__((address_space(1))) v4i_t* g_v4i_p;   // global b128 pointer
typedef __attribute__((address_space(3))) v4i_t* l_v4i_p;   // LDS b128 pointer
#endif

typedef float v4f_t __attribute__((ext_vector_type(4)));

// ---------------------------------------------------------------------------
// samples: [N*K, 257] gather-concat. One workgroup per node i (8 wave32s).
// Node row staged once in LDS via async global->LDS DMA (CDNA5 ASYNCcnt path),
// hoisted into VGPRs once per wave, then streamed out 64x with non-temporal
// coalesced dword stores (row start is only 4B-aligned -> dword pattern).
// ---------------------------------------------------------------------------
__global__ __launch_bounds__(256) void samples_kernel(const float* __restrict__ emb_in,
                                                      const float* __restrict__ sum_weights,
                                                      const int* __restrict__ idx_in,
                                                      float* __restrict__ samples) {
    __shared__ __attribute__((aligned(16))) float snode[FDIM];
    const int i   = blockIdx.x;
    const int tid = threadIdx.x;

    if (tid < 32) {
#ifdef HAVE_ASYNC_LDS
        // 32 lanes x 16B = 512B row copied by the async DMA path (no VGPR roundtrip)
        __builtin_amdgcn_global_load_async_to_lds_b128(
            (g_v4i_p)(emb_in + (size_t)i * FDIM + tid * 4),
            (l_v4i_p)(&snode[tid * 4]),
            0, 0);
#else
        #pragma unroll
        for (int k = 0; k < 4; ++k)
            snode[tid * 4 + k] = emb_in[(size_t)i * FDIM + tid * 4 + k];
#endif
    }
#ifdef HAVE_ASYNC_LDS
#  ifdef HAVE_WAIT_ASYNC
    __builtin_amdgcn_s_wait_asynccnt(0);
#  else
    asm volatile("s_wait_asynccnt 0" ::: "memory");
#  endif
#endif
    __syncthreads();

    const int wave = tid >> 5;
    const int lane = tid & 31;

    // Hoist node row into registers once per wave (stride-32 dword pattern)
    float nodev[4];
    #pragma unroll
    for (int k = 0; k < 4; ++k) nodev[k] = snode[lane + 32 * k];

    for (int j = wave; j < KNB; j += 8) {
        int c = (j < KM1) ? idx_in[i * KM1 + j] : i;
        float* row = samples + (size_t)(i * KNB + j) * ROWLEN;
        const float* nb = emb_in + (size_t)c * FDIM;   // L2-resident gather (4 MB table)
        #pragma unroll
        for (int k = 0; k < 4; ++k) {
            __builtin_nontemporal_store(nodev[k], row + lane + 32 * k);
            __builtin_nontemporal_store(nb[lane + 32 * k], row + FDIM + lane + 32 * k);
        }
        if (lane == 0)
            __builtin_nontemporal_store(sum_weights[c], row + 2 * FDIM);
    }
}

// ---------------------------------------------------------------------------
// adj fill: thread = (4 columns, 128-row block). Binary-search the carried
// value at block entry, then walk rows emitting one b128 NT store per row
// (512B per wave-instruction, fully coalesced).
// ---------------------------------------------------------------------------
#define RPB 128
#define CPT 4
__global__ __launch_bounds__(256) void adj_kernel(const int* __restrict__ offs,
                                                  const int* __restrict__ keys,
                                                  const float* __restrict__ vals,
                                                  float* __restrict__ adj) {
    const int c0 = (blockIdx.x * 256 + threadIdx.x) * CPT;
    const int r0 = blockIdx.y * RPB;
    const int t0 = r0 * KNB;

    int   off[CPT], n[CPT], ptr[CPT];
    float val[CPT];
    #pragma unroll
    for (int q = 0; q < CPT; ++q) {
        off[q] = offs[c0 + q];
        n[q]   = offs[c0 + q + 1] - off[q];
        int lo = 0, hi = n[q];
        while (lo < hi) {                       // count of events with row < r0
            int mid = (lo + hi) >> 1;
            if (keys[off[q] + mid] < t0) lo = mid + 1; else hi = mid;
        }
        ptr[q] = lo;
        val[q] = (lo > 0) ? vals[off[q] + lo - 1] : 0.0f;
    }

    for (int r = r0; r < r0 + RPB; ++r) {
        const int lim = (r + 1) * KNB;
        #pragma unroll
        for (int q = 0; q < CPT; ++q) {
            while (ptr[q] < n[q] && keys[off[q] + ptr[q]] < lim) {
                val[q] = vals[off[q] + ptr[q]];
                ++ptr[q];
            }
        }
        v4f_t v = { val[0], val[1], val[2], val[3] };
        __builtin_nontemporal_store(v, (v4f_t*)(adj + (size_t)r * N_NODES + c0));
    }
}

// ---------------------------------------------------------------------------
// Workspace kernels for the adj (sequential-scatter-with-carry) inversion
// ---------------------------------------------------------------------------

__global__ void zero_counts_kernel(int* __restrict__ counts, int* __restrict__ cursor) {
    int t = blockIdx.x * blockDim.x + threadIdx.x;
    if (t < N_NODES) { counts[t] = 0; cursor[t] = 0; }
}

__global__ void hist_kernel(const int* __restrict__ idx_in, int* __restrict__ counts) {
    int t = blockIdx.x * blockDim.x + threadIdx.x;   // 0 .. NK-1
    int i = t >> 6, j = t & 63;
    int c = (j < KM1) ? idx_in[i * KM1 + j] : i;     // self index appended last
    atomicAdd(&counts[c], 1);
}

// Single-block exclusive scan over 8192 counts -> offs[0..8192]
__global__ __launch_bounds__(1024) void scan_kernel(const int* __restrict__ counts,
                                                    int* __restrict__ offs) {
    __shared__ int part[1024];
    int t = threadIdx.x;
    int local[8];
    int s = 0;
    #pragma unroll
    for (int e = 0; e < 8; ++e) { local[e] = s; s += counts[t * 8 + e]; }
    part[t] = s;
    __syncthreads();
    for (int d = 1; d < 1024; d <<= 1) {
        int v = (t >= d) ? part[t - d] : 0;
        __syncthreads();
        part[t] += v;
        __syncthreads();
    }
    int base = (t > 0) ? part[t - 1] : 0;
    #pragma unroll
    for (int e = 0; e < 8; ++e) offs[t * 8 + e] = base + local[e];
    if (t == 1023) offs[N_NODES] = part[1023];
}

__global__ void scatter_kernel(const int* __restrict__ idx_in,
                               const float* __restrict__ edge_out,
                               const int* __restrict__ offs,
                               int* __restrict__ cursor,
                               int* __restrict__ keys,
                               float* __restrict__ vals) {
    int t = blockIdx.x * blockDim.x + threadIdx.x;   // order key == t == i*K + j
    int i = t >> 6, j = t & 63;
    int c = (j < KM1) ? idx_in[i * KM1 + j] : i;
    int pos = atomicAdd(&cursor[c], 1);
    int base = offs[c] + pos;
    keys[base] = t;
    vals[base] = edge_out[t];
}

// Per-column insertion sort (unique keys -> deterministic result). Lists are
// avg 64 long and fully L2-resident (4 MB total).
__global__ void sort_kernel(const int* __restrict__ offs,
                            int* __restrict__ keys,
                            float* __restrict__ vals) {
    int c = blockIdx.x * blockDim.x + threadIdx.x;
    if (c >= N_NODES) return;
    int off = offs[c];
    int n = offs[c + 1] - off;
    for (int a = 1; a < n; ++a) {
        int   kx = keys[off + a];
        float vx = vals[off + a];
        int b = a - 1;
        while (b >= 0 && keys[off + b] > kx) {
            keys[off + b + 1] = keys[off + b];
            vals[off + b + 1] = vals[off + b];
            --b;
        }
        keys[off + b + 1] = kx;
        vals[off + b + 1] = vx;
    }
}

// ---------------------------------------------------------------------------
extern "C" void kernel_launch(void* const* d_in, const int* in_sizes, int n_in,
                              void* d_out, int out_size, void* d_ws, size_t ws_size,
                              hipStream_t stream) {
    const float* emb_in      = (const float*)d_in[0];   // [N, F]
    const float* sum_weights = (const float*)d_in[1];   // [N]
    const int*   idx_in      = (const int*)d_in[2];     // [N, K-1]
    const float* edge_out    = (const float*)d_in[3];   // [N, K]

    float* out     = (float*)d_out;
    float* samples = out;                       // [N*K, 257]
    float* adj     = out + SAMPLES_ELEMS;       // [N, N]

    // Workspace layout (~4.2 MB total)
    char* ws = (char*)d_ws;
    int*   counts = (int*)(ws);                          //  8192 ints
    int*   cursor = (int*)(ws + 32768);                  //  8192 ints
    int*   offs   = (int*)(ws + 65536);                  //  8193 ints
    int*   keys   = (int*)(ws + 65536 + 33024);          //  NK ints   (@ 98560)
    float* vals   = (float*)(ws + 65536 + 33024 + 4UL * NK); // NK floats

    zero_counts_kernel<<<32, 256, 0, stream>>>(counts, cursor);
    hist_kernel<<<NK / 256, 256, 0, stream>>>(idx_in, counts);
    scan_kernel<<<1, 1024, 0, stream>>>(counts, offs);
    scatter_kernel<<<NK / 256, 256, 0, stream>>>(idx_in, edge_out, offs, cursor, keys, vals);
    sort_kernel<<<N_NODES / 256, 256, 0, stream>>>(offs, keys, vals);

    samples_kernel<<<N_NODES, 256, 0, stream>>>(emb_in, sum_weights, idx_in, samples);

    dim3 adj_grid(N_NODES / (256 * CPT), N_NODES / RPB);
    adj_kernel<<<adj_grid, 256, 0, stream>>>(offs, keys, vals, adj);
}